// Attention_4260607558337
// MI455X (gfx1250) — compile-verified
//
#include <hip/hip_runtime.h>
#include <hip/hip_bf16.h>

#define BATCH 32
#define SLEN  2048
#define DIM   1024
#define SCHUNK 128
#define NCHUNK (SLEN / SCHUNK)   // 16

typedef __attribute__((ext_vector_type(2))) float v2f;
typedef __attribute__((ext_vector_type(4))) float v4f;
typedef __attribute__((ext_vector_type(8))) float v8f;

// ---------------------------------------------------------------------------
// Kernel 1: dec_feature[32,1024] = dec_hidden[32,1024] @ W_dec^T + b_dec
// Full-f32 WMMA: V_WMMA_F32_16X16X4_F32, one 16x16 output tile per wave.
// Tiles: 2 (M) x 64 (N) = 128 waves. K loop: 1024/4 = 256 WMMAs per wave.
// A fragment (16x4): lane<16 -> M=lane, K={k,k+1}; lane>=16 -> M=lane-16, K={k+2,k+3}
// B fragment (4x16): B[k][n] = W_dec[n][k]; lane<16 -> N=lane, K={k,k+1}; hi lanes K={k+2,k+3}
// C/D (16x16 f32): VGPR r, lane L -> M = r + 8*(L>=16), N = L&15
// ---------------------------------------------------------------------------
__global__ __launch_bounds__(128) void decfeat_wmma_kernel(
    const float* __restrict__ dec_hidden,
    const float* __restrict__ W_dec,
    const float* __restrict__ b_dec,
    float* __restrict__ dec_feature) {
  const int waveId = (blockIdx.x * blockDim.x + threadIdx.x) >> 5;
  const int lane   = threadIdx.x & 31;
  const int mtile  = waveId & 1;        // 2 M-tiles
  const int ntile  = waveId >> 1;       // 64 N-tiles
  const int mrow   = mtile * 16 + (lane & 15);
  const int nrow   = ntile * 16 + (lane & 15);
  const int koff   = (lane >> 4) * 2;

  const float* aptr = dec_hidden + (size_t)mrow * DIM + koff;
  const float* bptr = W_dec      + (size_t)nrow * DIM + koff;

  v8f c = {};
  for (int k = 0; k < DIM; k += 4) {
    v2f a = *(const v2f*)(aptr + k);
    v2f b = *(const v2f*)(bptr + k);
    c = __builtin_amdgcn_wmma_f32_16x16x4_f32(
        /*neg_a=*/false, a, /*neg_b=*/false, b,
        /*c_mod=*/(short)0, c, /*reuse_a=*/false, /*reuse_b=*/false);
  }

  const int mbase = mtile * 16 + ((lane >> 4) << 3);
  const int n     = ntile * 16 + (lane & 15);
  const float bias = b_dec[n];
#pragma unroll
  for (int r = 0; r < 8; ++r) {
    dec_feature[(size_t)(mbase + r) * DIM + n] = c[r] + bias;
  }
}

// ---------------------------------------------------------------------------
// Kernel 2: score[b,s] = sum_d v[d]*tanh(enc_feature[b,s,d] + dec_feature[b,d]
//                                        + coverage[b,s]*w_cov[d])
// One wave32 per (b,s) row; NT b128 loads of the 256MB enc_feature stream.
// ---------------------------------------------------------------------------
__global__ __launch_bounds__(256) void score_kernel(
    const float* __restrict__ enc_feature,
    const float* __restrict__ dec_feature,
    const float* __restrict__ coverage,
    const float* __restrict__ v,
    const float* __restrict__ w_cov,
    float* __restrict__ score) {
  const int gwave = (blockIdx.x * blockDim.x + threadIdx.x) >> 5;
  const int lane  = threadIdx.x & 31;
  const int b = gwave >> 11;          // /SLEN
  const int s = gwave & (SLEN - 1);

  const float* ef = enc_feature + ((size_t)b * SLEN + s) * DIM;
  const float* df = dec_feature + (size_t)b * DIM;
  const float  cov = coverage[(size_t)b * SLEN + s];

  float acc = 0.f;
#pragma unroll
  for (int i = 0; i < DIM / (32 * 4); ++i) {
    const int d = (i * 32 + lane) * 4;
    v4f e  = __builtin_nontemporal_load((const v4f*)(ef + d));   // streamed once
    v4f dv = *(const v4f*)(df + d);                              // L2 resident
    v4f vv = *(const v4f*)(v + d);
    v4f wc = *(const v4f*)(w_cov + d);
    acc += vv.x * tanhf(e.x + dv.x + cov * wc.x);
    acc += vv.y * tanhf(e.y + dv.y + cov * wc.y);
    acc += vv.z * tanhf(e.z + dv.z + cov * wc.z);
    acc += vv.w * tanhf(e.w + dv.w + cov * wc.w);
  }
  // wave32 reduction
#pragma unroll
  for (int off = 16; off > 0; off >>= 1) acc += __shfl_xor(acc, off, 32);
  if (lane == 0) score[(size_t)b * SLEN + s] = acc;
}

// ---------------------------------------------------------------------------
// Kernel 3: per-batch softmax over S, mask, renormalize; coverage_new.
// One 256-thread block per batch; exponentials cached in LDS (8KB of 320KB).
// ---------------------------------------------------------------------------
__global__ __launch_bounds__(256) void softmax_kernel(
    const float* __restrict__ score,
    const float* __restrict__ enc_mask,
    const float* __restrict__ coverage,
    float* __restrict__ attn,
    float* __restrict__ cov_out) {
  __shared__ float e[SLEN];
  __shared__ float red[256];
  const int b = blockIdx.x;
  const int t = threadIdx.x;
  const float* srow = score + (size_t)b * SLEN;

  float lmax = -3.402823466e38f;
  for (int s = t; s < SLEN; s += 256) lmax = fmaxf(lmax, srow[s]);
  red[t] = lmax; __syncthreads();
  for (int off = 128; off > 0; off >>= 1) {
    if (t < off) red[t] = fmaxf(red[t], red[t + off]);
    __syncthreads();
  }
  const float m = red[0];
  __syncthreads();

  float lsum = 0.f;
  for (int s = t; s < SLEN; s += 256) {
    float x = __expf(srow[s] - m);
    e[s] = x;
    lsum += x;
  }
  red[t] = lsum; __syncthreads();
  for (int off = 128; off > 0; off >>= 1) {
    if (t < off) red[t] += red[t + off];
    __syncthreads();
  }
  const float invZ = 1.f / red[0];
  __syncthreads();

  float lsum2 = 0.f;
  for (int s = t; s < SLEN; s += 256) {
    float p = e[s] * invZ * enc_mask[(size_t)b * SLEN + s];
    e[s] = p;
    lsum2 += p;
  }
  red[t] = lsum2; __syncthreads();
  for (int off = 128; off > 0; off >>= 1) {
    if (t < off) red[t] += red[t + off];
    __syncthreads();
  }
  const float invZ2 = 1.f / red[0];
  __syncthreads();

  for (int s = t; s < SLEN; s += 256) {
    float a = e[s] * invZ2;
    attn[(size_t)b * SLEN + s] = a;
    cov_out[(size_t)b * SLEN + s] = coverage[(size_t)b * SLEN + s] + a;
  }
}

// ---------------------------------------------------------------------------
// Kernel 4: context partials. Block = (b, S-chunk of 128 rows); thread t owns
// d = 4t..4t+3 and accumulates over the chunk. NT b128 loads of enc_output.
// Deterministic two-phase sum (no float atomics -> bitwise reproducible).
// ---------------------------------------------------------------------------
__global__ __launch_bounds__(256) void context_partial_kernel(
    const float* __restrict__ enc_output,
    const float* __restrict__ attn,
    float* __restrict__ partials) {
  const int b     = blockIdx.x / NCHUNK;
  const int chunk = blockIdx.x % NCHUNK;
  const int d     = threadIdx.x * 4;

  const float* eo = enc_output + ((size_t)b * SLEN + (size_t)chunk * SCHUNK) * DIM + d;
  const float* at = attn + (size_t)b * SLEN + (size_t)chunk * SCHUNK;

  v4f acc = {0.f, 0.f, 0.f, 0.f};
  for (int s = 0; s < SCHUNK; ++s) {
    const float w = at[s];                                   // uniform -> scalar
    v4f x = __builtin_nontemporal_load((const v4f*)(eo + (size_t)s * DIM));
    acc.x += w * x.x; acc.y += w * x.y; acc.z += w * x.z; acc.w += w * x.w;
  }
  *(v4f*)(partials + ((size_t)(b * NCHUNK + chunk)) * DIM + d) = acc;
}

__global__ __launch_bounds__(256) void context_reduce_kernel(
    const float* __restrict__ partials,
    float* __restrict__ context) {
  const int idx = blockIdx.x * blockDim.x + threadIdx.x;   // B*DIM threads
  const int b = idx / DIM;
  const int d = idx % DIM;
  float acc = 0.f;
#pragma unroll
  for (int c = 0; c < NCHUNK; ++c)
    acc += partials[((size_t)(b * NCHUNK + c)) * DIM + d];
  context[idx] = acc;
}

// ---------------------------------------------------------------------------
// Launch
// ---------------------------------------------------------------------------
extern "C" void kernel_launch(void* const* d_in, const int* in_sizes, int n_in,
                              void* d_out, int out_size, void* d_ws, size_t ws_size,
                              hipStream_t stream) {
  const float* dec_hidden  = (const float*)d_in[0];
  const float* enc_output  = (const float*)d_in[1];
  const float* enc_feature = (const float*)d_in[2];
  const float* enc_mask    = (const float*)d_in[3];
  // d_in[4] = sec_attn: dead in the reference (overwritten), unused here.
  const float* coverage    = (const float*)d_in[5];
  const float* W_dec       = (const float*)d_in[6];
  const float* b_dec       = (const float*)d_in[7];
  const float* v           = (const float*)d_in[8];
  const float* w_cov       = (const float*)d_in[9];

  float* out      = (float*)d_out;
  float* context  = out;                        // [32,1024]
  float* attn     = out + BATCH * DIM;          // [32,2048]
  float* cov_out  = attn + BATCH * SLEN;        // [32,2048]

  float* ws          = (float*)d_ws;
  float* dec_feature = ws;                              // 32768 floats
  float* score       = dec_feature + BATCH * DIM;       // 65536 floats
  float* partials    = score + BATCH * SLEN;            // 32*16*1024 floats

  // 1) f32 WMMA GEMM: 128 waves = 32 blocks x 4 waves
  decfeat_wmma_kernel<<<32, 128, 0, stream>>>(dec_hidden, W_dec, b_dec, dec_feature);

  // 2) score: one wave per (b,s); 65536 waves / 8 per block = 8192 blocks
  score_kernel<<<(BATCH * SLEN) / 8, 256, 0, stream>>>(
      enc_feature, dec_feature, coverage, v, w_cov, score);

  // 3) softmax + mask renorm + coverage_new: one block per batch
  softmax_kernel<<<BATCH, 256, 0, stream>>>(score, enc_mask, coverage, attn, cov_out);

  // 4) context partials: 32 * 16 = 512 blocks
  context_partial_kernel<<<BATCH * NCHUNK, 256, 0, stream>>>(enc_output, attn, partials);

  // 5) reduce partials: 32768 threads
  context_reduce_kernel<<<(BATCH * DIM) / 256, 256, 0, stream>>>(partials, context);
}